// SegmentLayerNormalization_82781199663298
// MI455X (gfx1250) — compile-verified
//
#include <hip/hip_runtime.h>
#include <hip/hip_bf16.h>
#include <stdint.h>

// Segment LayerNorm, MI455X (gfx1250).
// One workgroup per segment. Segment rows are staged into LDS once via the
// CDNA5 async global->LDS DMA path (ASYNCcnt), so input is read from HBM
// exactly once. Input staging uses TH_LOAD_NT and output uses NT stores so
// the streamed 1 GB doesn't evict the L2 lines that oversized-segment
// fallbacks rely on. Roofline: ~1 GB HBM traffic -> ~44 us at 23.3 TB/s.

#define N_ROWS    262144
#define DIM       512
#define NSEG      4096
#define SEG_EPS   1e-5f
#define BLOCK     256
#define NWAVES    (BLOCK / 32)          // wave32: 8 waves per block
#define CAP_ROWS  72                    // 72 rows * 2KB = 147456 B dynamic LDS
#define SMEM_BYTES (CAP_ROWS * DIM * (int)sizeof(float))

typedef float v4f __attribute__((ext_vector_type(4)));

__global__ __launch_bounds__(BLOCK) void seg_ln_async_kernel(
    const float* __restrict__ x,        // [N_ROWS, DIM]
    const long long* __restrict__ ids,  // [N_ROWS], sorted
    const float* __restrict__ gain,     // [DIM]
    const float* __restrict__ bias,     // [DIM]
    float* __restrict__ out)            // [N_ROWS, DIM]
{
  extern __shared__ __align__(16) float stage[];   // dynamic: segment staging
  __shared__ float s_gain[DIM];
  __shared__ float s_bias[DIM];
  __shared__ float s_red[2 * NWAVES];
  __shared__ float s_stats[2];                     // mean, rstd

  const int tid = threadIdx.x;
  const int seg = blockIdx.x;

  // Cache gain/bias in LDS (2 elements per thread).
  for (int i = tid; i < DIM; i += BLOCK) {
    s_gain[i] = gain[i];
    s_bias[i] = bias[i];
  }

  // Row range of this segment via binary search (uniform -> SALU off ttmp9).
  const long long target = (long long)seg;
  int lo = 0, hi = N_ROWS;
  while (lo < hi) { int mid = (lo + hi) >> 1; if (ids[mid] <  target) lo = mid + 1; else hi = mid; }
  const int row_start = lo;
  hi = N_ROWS;
  while (lo < hi) { int mid = (lo + hi) >> 1; if (ids[mid] <= target) lo = mid + 1; else hi = mid; }
  const int row_end = lo;
  const int rows = row_end - row_start;

  __syncthreads();                 // gain/bias visible
  if (rows <= 0) return;           // uniform exit: empty segment

  const int chunks = rows * (DIM / 4);                    // 16-byte chunks
  const float* seg_base = x + (size_t)row_start * (size_t)DIM;
  const bool in_lds = (rows <= CAP_ROWS);                 // uniform per block

  float s = 0.0f, ss = 0.0f;

  if (in_lds) {
    // ---- Stage whole segment into LDS with async DMA (ASYNCcnt path) ----
    // Generic->LDS address is a low-32-bit truncation on AMDGPU.
    // NT hint: these lines are consumed from LDS, never re-read from cache.
    const unsigned lds_base = (unsigned)(uintptr_t)(const void*)stage;
    const uint64_t gbase    = (uint64_t)(uintptr_t)seg_base;
    for (int c = tid; c < chunks; c += BLOCK) {
      unsigned loff  = lds_base + ((unsigned)c << 4);
      uint64_t gaddr = gbase + ((uint64_t)(unsigned)c << 4);
      asm volatile("global_load_async_to_lds_b128 %0, %1, off th:TH_LOAD_NT"
                   :: "v"(loff), "v"(gaddr)
                   : "memory");
    }
    asm volatile("s_wait_asynccnt 0x0" ::: "memory");     // my wave's DMAs done
    __syncthreads();                                      // all waves' DMAs done

    const v4f* sp = (const v4f*)stage;
    for (int c = tid; c < chunks; c += BLOCK) {
      v4f v = sp[c];                                      // ds_load_b128
      s  += (v.x + v.y) + (v.z + v.w);
      ss = fmaf(v.x, v.x, ss); ss = fmaf(v.y, v.y, ss);
      ss = fmaf(v.z, v.z, ss); ss = fmaf(v.w, v.w, ss);
    }
  } else {
    // Oversized segment: stats pass straight from global (lands in L2;
    // deliberately regular-temporal so pass C re-reads hit L2).
    const v4f* gp = (const v4f*)seg_base;
    for (int c = tid; c < chunks; c += BLOCK) {
      v4f v = gp[c];
      s  += (v.x + v.y) + (v.z + v.w);
      ss = fmaf(v.x, v.x, ss); ss = fmaf(v.y, v.y, ss);
      ss = fmaf(v.z, v.z, ss); ss = fmaf(v.w, v.w, ss);
    }
  }

  // ---- Block reduction: wave32 shuffle tree, then cross-wave via LDS ----
  #pragma unroll
  for (int off = 16; off > 0; off >>= 1) {
    s  += __shfl_down(s,  off, 32);
    ss += __shfl_down(ss, off, 32);
  }
  const int wid = tid >> 5;
  if ((tid & 31) == 0) { s_red[wid] = s; s_red[NWAVES + wid] = ss; }
  __syncthreads();
  if (tid == 0) {
    float ts = 0.0f, tss = 0.0f;
    #pragma unroll
    for (int w = 0; w < NWAVES; ++w) { ts += s_red[w]; tss += s_red[NWAVES + w]; }
    const float cnt  = (float)rows * (float)DIM;
    const float mean = ts / cnt;
    const float var  = tss / cnt - mean * mean;           // biased variance
    s_stats[0] = mean;
    s_stats[1] = rsqrtf(var + SEG_EPS);
  }
  __syncthreads();

  const float mean = s_stats[0];
  const float rstd = s_stats[1];

  // ---- Normalize + affine, streamed out as non-temporal b128 stores ----
  v4f* op = (v4f*)(out + (size_t)row_start * (size_t)DIM);
  if (in_lds) {
    const v4f* sp = (const v4f*)stage;
    for (int c = tid; c < chunks; c += BLOCK) {
      v4f v = sp[c];
      const int col = (c & (DIM / 4 - 1)) << 2;
      v4f r;
      r.x = fmaf(s_gain[col + 0], (v.x - mean) * rstd, s_bias[col + 0]);
      r.y = fmaf(s_gain[col + 1], (v.y - mean) * rstd, s_bias[col + 1]);
      r.z = fmaf(s_gain[col + 2], (v.z - mean) * rstd, s_bias[col + 2]);
      r.w = fmaf(s_gain[col + 3], (v.w - mean) * rstd, s_bias[col + 3]);
      __builtin_nontemporal_store(r, op + c);
    }
  } else {
    const v4f* gp = (const v4f*)seg_base;                 // re-read: L2 hit
    for (int c = tid; c < chunks; c += BLOCK) {
      v4f v = gp[c];
      const int col = (c & (DIM / 4 - 1)) << 2;
      v4f r;
      r.x = fmaf(s_gain[col + 0], (v.x - mean) * rstd, s_bias[col + 0]);
      r.y = fmaf(s_gain[col + 1], (v.y - mean) * rstd, s_bias[col + 1]);
      r.z = fmaf(s_gain[col + 2], (v.z - mean) * rstd, s_bias[col + 2]);
      r.w = fmaf(s_gain[col + 3], (v.w - mean) * rstd, s_bias[col + 3]);
      __builtin_nontemporal_store(r, op + c);
    }
  }
}

extern "C" void kernel_launch(void* const* d_in, const int* in_sizes, int n_in,
                              void* d_out, int out_size, void* d_ws, size_t ws_size,
                              hipStream_t stream) {
  const float*     x    = (const float*)d_in[0];
  const long long* ids  = (const long long*)d_in[1];
  const float*     gain = (const float*)d_in[2];
  const float*     bias = (const float*)d_in[3];
  float*           out  = (float*)d_out;

  // Allow >64KB dynamic LDS (no-op / harmless if the runtime doesn't need it).
  (void)hipFuncSetAttribute(reinterpret_cast<const void*>(&seg_ln_async_kernel),
                            hipFuncAttributeMaxDynamicSharedMemorySize, SMEM_BYTES);

  seg_ln_async_kernel<<<NSEG, BLOCK, SMEM_BYTES, stream>>>(x, ids, gain, bias, out);
}